// II_MVF_3676492005814
// MI455X (gfx1250) — compile-verified
//
#include <hip/hip_runtime.h>

typedef float v2f __attribute__((ext_vector_type(2)));
typedef float v8f __attribute__((ext_vector_type(8)));

#define N_    16
#define DIM_  2048
#define RF_   256
#define HW_   128
#define PN_   64
#define TOPK_ 10
#define AFF_  (1.0f / 262144.0f)   // c_avg*s_avg = (1/2048)*(1/128), exact

__device__ __forceinline__ v8f wmma4(v2f a, v2f b, v8f c) {
    // V_WMMA_F32_16X16X4_F32 : D = A(16x4 f32) x B(4x16 f32) + C(16x16 f32)
    return __builtin_amdgcn_wmma_f32_16x16x4_f32(
        /*neg_a=*/false, a, /*neg_b=*/false, b,
        /*c_mod=*/(short)0, c, /*reuse_a=*/false, /*reuse_b=*/false);
}

// ---------------------------------------------------------------------------
// Kernel 1: squared euclidean distance  dist[n,p] = sum((x_n - p_p)^2)
// (monotone-identical to the reference addmm formula; ordering is all we need)
// ---------------------------------------------------------------------------
__global__ void k_dist(const float* __restrict__ x, const float* __restrict__ p,
                       float* __restrict__ dist) {
    const int pp = blockIdx.x, n = blockIdx.y;
    const float* xv = x + (size_t)n  * (DIM_ * HW_);
    const float* pv = p + (size_t)pp * (DIM_ * HW_);
    float s = 0.0f;
    for (int i = threadIdx.x; i < DIM_ * HW_; i += 256) {
        float d = xv[i] - pv[i];
        s += d * d;
    }
    __shared__ float red[256];
    red[threadIdx.x] = s;
    __syncthreads();
    for (int off = 128; off > 0; off >>= 1) {
        if (threadIdx.x < off) red[threadIdx.x] += red[threadIdx.x + off];
        __syncthreads();
    }
    if (threadIdx.x == 0) dist[n * PN_ + pp] = red[0];
}

// ---------------------------------------------------------------------------
// Kernel 2: stable top-10 smallest per sample (matches argsort[:, :TOPK])
// ---------------------------------------------------------------------------
__global__ void k_topk(const float* __restrict__ dist, int* __restrict__ idx) {
    const int n = threadIdx.x;
    if (n >= N_) return;
    float d[PN_];
    bool used[PN_];
    for (int j = 0; j < PN_; ++j) { d[j] = dist[n * PN_ + j]; used[j] = false; }
    for (int t = 0; t < TOPK_; ++t) {
        int bi = 0; float bv = 3.4e38f;
        for (int j = 0; j < PN_; ++j)
            if (!used[j] && d[j] < bv) { bv = d[j]; bi = j; }
        used[bi] = true;
        idx[n * TOPK_ + t] = bi;
    }
}

// ---------------------------------------------------------------------------
// Kernel 3: xp[p, r, pix] = sum_c w_rf[r,c] * proto[p,c,pix]
// grid (16 rBlocks, 64 protos), 256 thr = 8 waves; wave w -> pixels [16w,16w+16)
// ---------------------------------------------------------------------------
__global__ void k_refine(const float* __restrict__ wrf, const float* __restrict__ proto,
                         float* __restrict__ xp) {
    const int rb = blockIdx.x, p = blockIdx.y;
    const int lane = threadIdx.x & 31, wave = threadIdx.x >> 5;
    const int mrow = lane & 15, khalf = lane >> 4;
    const int pix = wave * 16 + mrow;
    const float* Ap = wrf   + (size_t)(rb * 16 + mrow) * DIM_ + 2 * khalf;
    const float* Bp = proto + (size_t)p * (DIM_ * HW_) + (size_t)(2 * khalf) * HW_ + pix;
    v8f c = {};
#pragma unroll 4
    for (int k = 0; k < DIM_; k += 4) {
        v2f a = *(const v2f*)(Ap + k);
        v2f b; b.x = Bp[(size_t)k * HW_]; b.y = Bp[(size_t)k * HW_ + HW_];
        c = wmma4(a, b, c);
    }
    float* op = xp + ((size_t)p * RF_ + rb * 16) * HW_;
#pragma unroll
    for (int i = 0; i < 8; ++i) {
        int r = i + 8 * khalf;
        op[(size_t)r * HW_ + pix] = c[i];
    }
}

// ---------------------------------------------------------------------------
// Kernel 4: mvl[n,o,pix] = sum_t sum_r w_rf_mv[o, t*256+r] * xp[idx[n,t], r, pix]
// fused calibration (constant blend) + ReLU epilogue.
// ---------------------------------------------------------------------------
__global__ void k_mvl(const float* __restrict__ wrfmv, const float* __restrict__ xp,
                      const int* __restrict__ idx, const float* __restrict__ x,
                      float* __restrict__ mvl) {
    const int ob = blockIdx.x, n = blockIdx.y;
    const int lane = threadIdx.x & 31, wave = threadIdx.x >> 5;
    const int mrow = lane & 15, khalf = lane >> 4;
    const int pix = wave * 16 + mrow;
    __shared__ int sidx[TOPK_];
    if (threadIdx.x < TOPK_) sidx[threadIdx.x] = idx[n * TOPK_ + threadIdx.x];
    __syncthreads();
    const float* Ab = wrfmv + (size_t)(ob * 16 + mrow) * (TOPK_ * RF_) + 2 * khalf;
    v8f c = {};
    for (int t = 0; t < TOPK_; ++t) {
        const float* Ap = Ab + t * RF_;
        const float* Bp = xp + (size_t)sidx[t] * (RF_ * HW_) + (size_t)(2 * khalf) * HW_ + pix;
#pragma unroll 4
        for (int k = 0; k < RF_; k += 4) {
            v2f a = *(const v2f*)(Ap + k);
            v2f b; b.x = Bp[(size_t)k * HW_]; b.y = Bp[(size_t)k * HW_ + HW_];
            c = wmma4(a, b, c);
        }
    }
    const float* xb = x   + ((size_t)n * DIM_ + ob * 16) * HW_;
    float*       op = mvl + ((size_t)n * DIM_ + ob * 16) * HW_;
#pragma unroll
    for (int i = 0; i < 8; ++i) {
        int r = i + 8 * khalf;
        float v = 0.5f * xb[(size_t)r * HW_ + pix] + (1.0f - AFF_) * c[i];
        op[(size_t)r * HW_ + pix] = fmaxf(v, 0.0f);
    }
}

// ---------------------------------------------------------------------------
// Kernel 5: mv_a = instance_norm(w_rm_s x mvl). Block owns 16 channels x all
// 128 pixels -> instance-norm reduction is block-local (LDS).
// ---------------------------------------------------------------------------
__global__ void k_rms(const float* __restrict__ wrm, const float* __restrict__ mvl,
                      float* __restrict__ mva) {
    const int ob = blockIdx.x, n = blockIdx.y;
    const int lane = threadIdx.x & 31, wave = threadIdx.x >> 5;
    const int mrow = lane & 15, khalf = lane >> 4;
    const int pix = wave * 16 + mrow;
    const float* Ap = wrm + (size_t)(ob * 16 + mrow) * DIM_ + 2 * khalf;
    const float* Bp = mvl + (size_t)n * (DIM_ * HW_) + (size_t)(2 * khalf) * HW_ + pix;
    v8f c = {};
#pragma unroll 4
    for (int k = 0; k < DIM_; k += 4) {
        v2f a = *(const v2f*)(Ap + k);
        v2f b; b.x = Bp[(size_t)k * HW_]; b.y = Bp[(size_t)k * HW_ + HW_];
        c = wmma4(a, b, c);
    }
    __shared__ float tile[16 * HW_];
    __shared__ float smean[16], srstd[16];
#pragma unroll
    for (int i = 0; i < 8; ++i) {
        int r = i + 8 * khalf;
        tile[r * HW_ + pix] = c[i];
    }
    __syncthreads();
    if (threadIdx.x < 16) {
        const float* rowp = tile + threadIdx.x * HW_;
        float s = 0.0f, s2 = 0.0f;
        for (int j = 0; j < HW_; ++j) { float v = rowp[j]; s += v; s2 += v * v; }
        float m = s * (1.0f / HW_);
        float var = s2 * (1.0f / HW_) - m * m;
        smean[threadIdx.x] = m;
        srstd[threadIdx.x] = rsqrtf(var + 1e-5f);
    }
    __syncthreads();
    float* op = mva + ((size_t)n * DIM_ + ob * 16) * HW_;
    for (int i = threadIdx.x; i < 16 * HW_; i += 256) {
        int r = i >> 7;
        op[i] = (tile[i] - smean[r]) * srstd[r];
    }
}

// ---------------------------------------------------------------------------
// Kernel 6: out = x + x * sigmoid(w_fuse x mv_a)
// ---------------------------------------------------------------------------
__global__ void k_fuse(const float* __restrict__ wf, const float* __restrict__ mva,
                       const float* __restrict__ x, float* __restrict__ out) {
    const int ob = blockIdx.x, n = blockIdx.y;
    const int lane = threadIdx.x & 31, wave = threadIdx.x >> 5;
    const int mrow = lane & 15, khalf = lane >> 4;
    const int pix = wave * 16 + mrow;
    const float* Ap = wf  + (size_t)(ob * 16 + mrow) * DIM_ + 2 * khalf;
    const float* Bp = mva + (size_t)n * (DIM_ * HW_) + (size_t)(2 * khalf) * HW_ + pix;
    v8f c = {};
#pragma unroll 4
    for (int k = 0; k < DIM_; k += 4) {
        v2f a = *(const v2f*)(Ap + k);
        v2f b; b.x = Bp[(size_t)k * HW_]; b.y = Bp[(size_t)k * HW_ + HW_];
        c = wmma4(a, b, c);
    }
    const float* xb = x   + ((size_t)n * DIM_ + ob * 16) * HW_;
    float*       op = out + ((size_t)n * DIM_ + ob * 16) * HW_;
#pragma unroll
    for (int i = 0; i < 8; ++i) {
        int r = i + 8 * khalf;
        float s = 1.0f / (1.0f + expf(-c[i]));
        float xv = xb[(size_t)r * HW_ + pix];
        op[(size_t)r * HW_ + pix] = xv + xv * s;
    }
}

extern "C" void kernel_launch(void* const* d_in, const int* in_sizes, int n_in,
                              void* d_out, int out_size, void* d_ws, size_t ws_size,
                              hipStream_t stream) {
    const float* x     = (const float*)d_in[0];   // [16, 2048, 16, 8]
    const float* proto = (const float*)d_in[1];   // [64, 2048, 16, 8]
    const float* wrf   = (const float*)d_in[2];   // [256, 2048]
    const float* wrfmv = (const float*)d_in[3];   // [2048, 2560]
    const float* wrms  = (const float*)d_in[4];   // [2048, 2048]
    const float* wfuse = (const float*)d_in[5];   // [2048, 2048]
    float* out = (float*)d_out;                   // [16, 2048, 16, 8]

    char* ws = (char*)d_ws;
    float* dist = (float*)ws;                                      // 1024 f
    int*   idx  = (int*)(ws + 4096);                               // 160 i32
    float* xp   = (float*)(ws + 8192);                             // 64*256*128 f (8 MB)
    float* mvl  = (float*)(ws + 8192 + (size_t)8 * 1024 * 1024);   // 16*2048*128 f (16 MB)
    float* mva  = (float*)(ws + 8192 + (size_t)24 * 1024 * 1024);  // 16*2048*128 f (16 MB)

    k_dist  <<<dim3(PN_, N_),        256, 0, stream>>>(x, proto, dist);
    k_topk  <<<1,                     32, 0, stream>>>(dist, idx);
    k_refine<<<dim3(RF_ / 16, PN_),  256, 0, stream>>>(wrf, proto, xp);
    k_mvl   <<<dim3(DIM_ / 16, N_),  256, 0, stream>>>(wrfmv, xp, idx, x, mvl);
    k_rms   <<<dim3(DIM_ / 16, N_),  256, 0, stream>>>(wrms, mvl, mva);
    k_fuse  <<<dim3(DIM_ / 16, N_),  256, 0, stream>>>(wfuse, mva, x, out);
}